// EAMForceCUDA_11854109737006
// MI455X (gfx1250) — compile-verified
//
#include <hip/hip_runtime.h>
#include <hip/hip_bf16.h>
#include <stdint.h>

// ---------------- problem constants (match reference) ----------------
#define N_ATOMS   100000
#define N_EDGES   3200000
#define N_ELEM    3
#define N_R       5000
#define N_RHO     2000

__device__ __constant__ float kBOX      = 6.0f;
__device__ __constant__ float kINV_BOX  = 1.0f / 6.0f;
__device__ __constant__ float kCUTOFF   = 6.0f;
__device__ __constant__ float kR_MAX    = 6.5f;
__device__ __constant__ float kHR       = 6.5f / 4999.0f;       // r-grid spacing
__device__ __constant__ float kINV_HR   = 4999.0f / 6.5f;
__device__ __constant__ float kRHO_LO   = -60.0f;
__device__ __constant__ float kHE       = 120.0f / 1999.0f;     // embed-grid spacing
__device__ __constant__ float kINV_HE   = 1999.0f / 120.0f;

#define M_R    (N_R - 1)      // 4999 spline intervals on r-grid
#define M_RHO  (N_RHO - 1)    // 1999 spline intervals on rho-grid
#define TAB_ENTRIES (M_R * N_ELEM)          // 14997 float4 records
#define TAB_BYTES   (TAB_ENTRIES * 16)      // 239,952 B  (< 320 KB WGP LDS)

// output layout: [0]=total_E, [1..300000]=forces, [300001]=emb_E, [300002]=pair_E
#define OUT_FORCES 1
#define OUT_EMB    (1 + 3 * N_ATOMS)
#define OUT_PAIR   (2 + 3 * N_ATOMS)
#define OUT_TOTAL  (3 + 3 * N_ATOMS)

#define AS1 __attribute__((address_space(1)))
#define AS3 __attribute__((address_space(3)))

typedef int v4i __attribute__((ext_vector_type(4)));

// ---------------- CDNA5 async global -> LDS copy (16B per lane) ----------------
__device__ __forceinline__ void async_copy16(const void* gsrc, void* ldst) {
#if __has_builtin(__builtin_amdgcn_global_load_async_to_lds_b128)
  __builtin_amdgcn_global_load_async_to_lds_b128(
      (AS1 v4i*)(uintptr_t)gsrc, (AS3 v4i*)ldst, 0, 0);
#else
  unsigned lofs = (unsigned)(uintptr_t)(AS3 char*)ldst;        // 32-bit LDS byte addr
  unsigned long long ga = (unsigned long long)(uintptr_t)gsrc; // 64-bit global addr
  asm volatile("global_load_async_to_lds_b128 %0, %1, off"
               :: "v"(lofs), "v"(ga) : "memory");
#endif
}

__device__ __forceinline__ void wait_async0() {
#if __has_builtin(__builtin_amdgcn_s_wait_asynccnt)
  __builtin_amdgcn_s_wait_asynccnt(0);
#else
  asm volatile("s_wait_asynccnt 0" ::: "memory");
#endif
}

__device__ __forceinline__ float wave_sum(float v) {
  #pragma unroll
  for (int m = 16; m > 0; m >>= 1) v += __shfl_xor(v, m, 32);
  return v;
}

// uniform-grid locate on r-grid: clamp + interval index + dx
__device__ __forceinline__ void locate_r(float r, int& idx, float& dx) {
  float vc = fminf(fmaxf(r, 0.0f), kR_MAX);
  int i = (int)(vc * kINV_HR);
  idx = i < (M_R - 1) ? i : (M_R - 1);
  dx = vc - (float)idx * kHR;
}

// ---------------- kernel 0: zero accumulators ----------------
__global__ void k_zero(float* __restrict__ out, float* __restrict__ rho,
                       float* __restrict__ dF) {
  int i = blockIdx.x * blockDim.x + threadIdx.x;
  int stride = gridDim.x * blockDim.x;
  for (int k = i; k < OUT_TOTAL; k += stride) out[k] = 0.0f;
  for (int k = i; k < N_ATOMS; k += stride) { rho[k] = 0.0f; dF[k] = 0.0f; }
}

// ---------------- kernel 1: relayout spline tables into (idx, elem, coef4) ----------------
// densC[i*3+e]         = (c0,c1,c2,c3) of density spline, element e, interval i
// ddC  [i*3+e]         = (d0,d1,d2,0)  of density deriv spline
// pairC[(i*3+a)*3+b]   = (c0,c1,c2,c3) of pair spline (a,b)
// pdC  [(i*3+a)*3+b]   = (d0,d1,d2,0)  of pair deriv spline
__global__ void k_relayout(const float* __restrict__ dsc, const float* __restrict__ ddc,
                           const float* __restrict__ psc, const float* __restrict__ pdc,
                           float4* __restrict__ densC, float4* __restrict__ ddC,
                           float4* __restrict__ pairC, float4* __restrict__ pdC) {
  int i = blockIdx.x * blockDim.x + threadIdx.x;
  if (i >= M_R) return;
  #pragma unroll
  for (int e = 0; e < N_ELEM; ++e) {
    densC[i * 3 + e] = make_float4(dsc[(e * 4 + 0) * M_R + i], dsc[(e * 4 + 1) * M_R + i],
                                   dsc[(e * 4 + 2) * M_R + i], dsc[(e * 4 + 3) * M_R + i]);
    ddC[i * 3 + e]   = make_float4(ddc[(e * 3 + 0) * M_R + i], ddc[(e * 3 + 1) * M_R + i],
                                   ddc[(e * 3 + 2) * M_R + i], 0.0f);
  }
  #pragma unroll
  for (int a = 0; a < N_ELEM; ++a)
    #pragma unroll
    for (int b = 0; b < N_ELEM; ++b) {
      const float* p = psc + (size_t)((a * 3 + b) * 4) * M_R;
      pairC[(i * 3 + a) * 3 + b] =
          make_float4(p[0 * M_R + i], p[1 * M_R + i], p[2 * M_R + i], p[3 * M_R + i]);
      const float* q = pdc + (size_t)((a * 3 + b) * 3) * M_R;
      pdC[(i * 3 + a) * 3 + b] =
          make_float4(q[0 * M_R + i], q[1 * M_R + i], q[2 * M_R + i], 0.0f);
    }
}

// ---------------- kernel 2: edge pass 1 (rho accumulation + pair energy) ----------------
__global__ void k_edge1(const float* __restrict__ coords, const int* __restrict__ ei,
                        const int* __restrict__ types, const float4* __restrict__ densC,
                        const float4* __restrict__ pairC, float* __restrict__ rho,
                        float* __restrict__ pairAcc) {
  extern __shared__ float4 sTab[];  // 14997 float4 = 240KB density coeff table
  for (int i = threadIdx.x; i < TAB_ENTRIES; i += blockDim.x)
    async_copy16(&densC[i], &sTab[i]);
  wait_async0();
  __syncthreads();

  float phiSum = 0.0f;
  int tid = blockIdx.x * blockDim.x + threadIdx.x;
  int stride = gridDim.x * blockDim.x;
  for (int e = tid; e < N_EDGES; e += stride) {
    int row = ei[e];
    int col = ei[N_EDGES + e];
    float dx0 = coords[3 * row + 0] - coords[3 * col + 0];
    float dy0 = coords[3 * row + 1] - coords[3 * col + 1];
    float dz0 = coords[3 * row + 2] - coords[3 * col + 2];
    dx0 -= rintf(dx0 * kINV_BOX) * kBOX;
    dy0 -= rintf(dy0 * kINV_BOX) * kBOX;
    dz0 -= rintf(dz0 * kINV_BOX) * kBOX;
    float r = sqrtf(dx0 * dx0 + dy0 * dy0 + dz0 * dz0);
    bool in = r < kCUTOFF;
    int idx; float dxr;
    locate_r(r, idx, dxr);
    int it = types[row];
    int jt = types[col];
    float4 dc = sTab[idx * 3 + jt];   // LDS hit (async-staged)
    float dens = dc.w + dxr * (dc.z + dxr * (dc.y + dxr * dc.x));
    if (in) atomicAdd(&rho[row], dens);
    float4 pc = pairC[(idx * 3 + it) * 3 + jt];  // L2-resident b128 gather
    float phi = pc.w + dxr * (pc.z + dxr * (pc.y + dxr * pc.x));
    phiSum += in ? phi : 0.0f;
  }
  float w = wave_sum(phiSum);
  if ((threadIdx.x & 31) == 0) atomicAdd(pairAcc, 0.5f * w);
}

// ---------------- kernel 3: per-atom embedding energy + dF/drho ----------------
__global__ void k_atom(const float* __restrict__ rho, const int* __restrict__ types,
                       const float* __restrict__ embC, const float* __restrict__ embD,
                       float* __restrict__ dF, float* __restrict__ embAcc) {
  float embSum = 0.0f;
  int tid = blockIdx.x * blockDim.x + threadIdx.x;
  int stride = gridDim.x * blockDim.x;
  for (int a = tid; a < N_ATOMS; a += stride) {
    int t = types[a];
    float v = rho[a];
    float vc = fminf(fmaxf(v, kRHO_LO), -kRHO_LO);
    int i = (int)((vc - kRHO_LO) * kINV_HE);
    int i2 = i < (M_RHO - 1) ? i : (M_RHO - 1);
    float dxe = vc - (kRHO_LO + (float)i2 * kHE);
    const float* c = embC + (size_t)t * 4 * M_RHO;
    float emb = c[3 * M_RHO + i2] +
                dxe * (c[2 * M_RHO + i2] + dxe * (c[1 * M_RHO + i2] + dxe * c[0 * M_RHO + i2]));
    const float* cd = embD + (size_t)t * 3 * M_RHO;
    float vd = cd[2 * M_RHO + i2] + dxe * (cd[1 * M_RHO + i2] + dxe * cd[0 * M_RHO + i2]);
    dF[a] = vd;
    embSum += emb;
  }
  float w = wave_sum(embSum);
  if ((threadIdx.x & 31) == 0) atomicAdd(embAcc, w);
}

// ---------------- kernel 4: edge pass 2 (forces) ----------------
__global__ void k_edge2(const float* __restrict__ coords, const int* __restrict__ ei,
                        const int* __restrict__ types, const float4* __restrict__ ddC,
                        const float4* __restrict__ pdC, const float* __restrict__ dF,
                        float* __restrict__ forces) {
  extern __shared__ float4 sTab[];  // density-deriv table in LDS
  for (int i = threadIdx.x; i < TAB_ENTRIES; i += blockDim.x)
    async_copy16(&ddC[i], &sTab[i]);
  wait_async0();
  __syncthreads();

  int tid = blockIdx.x * blockDim.x + threadIdx.x;
  int stride = gridDim.x * blockDim.x;
  for (int e = tid; e < N_EDGES; e += stride) {
    int row = ei[e];
    int col = ei[N_EDGES + e];
    float dx0 = coords[3 * row + 0] - coords[3 * col + 0];
    float dy0 = coords[3 * row + 1] - coords[3 * col + 1];
    float dz0 = coords[3 * row + 2] - coords[3 * col + 2];
    dx0 -= rintf(dx0 * kINV_BOX) * kBOX;
    dy0 -= rintf(dy0 * kINV_BOX) * kBOX;
    dz0 -= rintf(dz0 * kINV_BOX) * kBOX;
    float r = sqrtf(dx0 * dx0 + dy0 * dy0 + dz0 * dz0);
    if (!(r < kCUTOFF)) continue;  // mask==0 contributes nothing
    int idx; float dxr;
    locate_r(r, idx, dxr);
    int it = types[row];
    int jt = types[col];
    float4 dj = sTab[idx * 3 + jt];
    float rho_p_j = dj.z + dxr * (dj.y + dxr * dj.x);
    float4 di = sTab[idx * 3 + it];
    float rho_p_i = di.z + dxr * (di.y + dxr * di.x);
    float4 pd = pdC[(idx * 3 + it) * 3 + jt];
    float phi_p = pd.z + dxr * (pd.y + dxr * pd.x);
    float fsc = dF[row] * rho_p_j + dF[col] * rho_p_i + phi_p;
    float s = -fsc / fmaxf(r, 1e-12f);
    atomicAdd(&forces[3 * row + 0], s * dx0);
    atomicAdd(&forces[3 * row + 1], s * dy0);
    atomicAdd(&forces[3 * row + 2], s * dz0);
  }
}

// ---------------- kernel 5: total energy ----------------
__global__ void k_final(float* __restrict__ out) {
  if (threadIdx.x == 0 && blockIdx.x == 0)
    out[0] = out[OUT_EMB] + out[OUT_PAIR];
}

// ---------------- host launcher ----------------
extern "C" void kernel_launch(void* const* d_in, const int* in_sizes, int n_in,
                              void* d_out, int out_size, void* d_ws, size_t ws_size,
                              hipStream_t stream) {
  const float* coords = (const float*)d_in[0];
  const int*   ei     = (const int*)d_in[1];
  const int*   types  = (const int*)d_in[2];
  // d_in[3] = spline_r_x (uniform, derived analytically), d_in[6] = embed_x (uniform)
  const float* dsc = (const float*)d_in[4];
  const float* ddc = (const float*)d_in[5];
  const float* embC = (const float*)d_in[7];
  const float* embD = (const float*)d_in[8];
  const float* psc = (const float*)d_in[9];
  const float* pdc = (const float*)d_in[10];
  float* out = (float*)d_out;

  // workspace layout (16B aligned chunks)
  char* ws = (char*)d_ws;
  float4* densC = (float4*)(ws);                               // 239,952 B
  float4* ddC   = (float4*)(ws + (size_t)TAB_BYTES);           // 239,952 B
  float4* pairC = (float4*)(ws + (size_t)TAB_BYTES * 2);       // 719,856 B
  float4* pdC   = (float4*)(ws + (size_t)TAB_BYTES * 2 + (size_t)M_R * 9 * 16);
  float*  rho   = (float*)(ws + (size_t)TAB_BYTES * 2 + (size_t)M_R * 9 * 16 * 2);
  float*  dF    = rho + N_ATOMS;

  k_zero<<<1024, 256, 0, stream>>>(out, rho, dF);
  k_relayout<<<(M_R + 255) / 256, 256, 0, stream>>>(dsc, ddc, psc, pdc,
                                                    densC, ddC, pairC, pdC);
  // 1024-thread blocks: 32 waves/WGP even with 240KB LDS staged per block
  k_edge1<<<256, 1024, TAB_BYTES, stream>>>(coords, ei, types, densC, pairC,
                                            rho, out + OUT_PAIR);
  k_atom<<<(N_ATOMS + 255) / 256, 256, 0, stream>>>(rho, types, embC, embD,
                                                    dF, out + OUT_EMB);
  k_edge2<<<256, 1024, TAB_BYTES, stream>>>(coords, ei, types, ddC, pdC, dF,
                                            out + OUT_FORCES);
  k_final<<<1, 32, 0, stream>>>(out);
}